// transforms_atkinson_dithering_33956011442779
// MI455X (gfx1250) — compile-verified
//
#include <hip/hip_runtime.h>
#include <cstdint>

// Atkinson dithering, 16 x 2048 x 2048 fp32, 8x8 independent blocks.
// Strategy: 1 workgroup (256 thr) = one contiguous 64 KiB strip (8 rows x 2048
// cols) staged into LDS via gfx1250 async-to-LDS engine; 1 thread = one 8x8
// block, fully register-resident sequential error diffusion.

#define BS   8
#define IMW  2048
#define IMH  2048
#define NBX  (IMW / BS)          // 256 blocks per strip = threads per WG
#define NBY  (IMH / BS)          // 256 strips per image
#define NIMG 16
#define STRIP_FLOATS (BS * IMW)  // 16384 floats
#define STRIP_BYTES  (STRIP_FLOATS * 4)  // 65536 B
#define ASYNC_ITERS  (STRIP_BYTES / (NBX * 16))  // 16 x b128 per thread

__global__ __launch_bounds__(NBX)
void atkinson_dither_kernel(const float* __restrict__ in, float* __restrict__ out) {
    __shared__ float smem[STRIP_FLOATS];   // 64 KiB strip

    const unsigned wg = blockIdx.x;        // 0 .. NIMG*NBY-1
    const unsigned b  = wg / NBY;          // image index
    const unsigned by = wg % NBY;          // block-row (strip) index
    const unsigned t  = threadIdx.x;       // block-col index bx = t

    // Strip is 8 consecutive image rows -> contiguous 64 KiB in global memory.
    const size_t stripBase = ((size_t)b * IMH + (size_t)by * BS) * (size_t)IMW;

    // ---- async stage: global -> LDS (gfx1250 async-to-LDS path) ----------
    // Flat shared pointer low 32 bits == workgroup-relative LDS byte address
    // (CDNA5 ISA: LDS address is the low 32 bits; async dst is LDS_BASE-relative).
    const unsigned long long sbase =
        (unsigned long long)(uintptr_t)(in + stripBase);
    const unsigned ldsbase = (unsigned)(uintptr_t)(&smem[0]);

    #pragma unroll
    for (int i = 0; i < ASYNC_ITERS; ++i) {
        // wave-contiguous: 32 lanes x 16 B = 512 B per issue
        unsigned boff = (unsigned)(i * NBX + (int)t) * 16u;
        unsigned ldst = ldsbase + boff;
        asm volatile("global_load_async_to_lds_b128 %0, %1, %2"
                     :: "v"(ldst), "v"(boff), "s"(sbase)
                     : "memory");
    }
    asm volatile("s_wait_asynccnt 0" ::: "memory");  // our wave's transfers done
    __syncthreads();                                  // everyone's transfers done

    // ---- pull 8x8 block into registers, scale to [0,255] -----------------
    float blk[BS][BS];
    #pragma unroll
    for (int y = 0; y < BS; ++y) {
        const float4 lo = *reinterpret_cast<const float4*>(&smem[y * IMW + t * BS]);
        const float4 hi = *reinterpret_cast<const float4*>(&smem[y * IMW + t * BS + 4]);
        blk[y][0] = lo.x * 255.0f;  blk[y][1] = lo.y * 255.0f;
        blk[y][2] = lo.z * 255.0f;  blk[y][3] = lo.w * 255.0f;
        blk[y][4] = hi.x * 255.0f;  blk[y][5] = hi.y * 255.0f;
        blk[y][6] = hi.z * 255.0f;  blk[y][7] = hi.w * 255.0f;
    }

    // ---- sequential Atkinson error diffusion (fully unrolled, all-VGPR) ---
    // Taps (0,1)(0,2)(1,-1)(1,0)(1,1)(2,0) are strictly later in raster
    // order, so blk[y][x] can be overwritten with the final 0/1 output.
    #pragma unroll
    for (int y = 0; y < BS; ++y) {
        #pragma unroll
        for (int x = 0; x < BS; ++x) {
            const float old = blk[y][x];
            const bool  hit = old > 127.0f;
            const float err = (old - (hit ? 255.0f : 0.0f)) * 0.125f; // exact *2^-3
            blk[y][x] = hit ? 1.0f : 0.0f;   // == (255 or 0) / 255
            if (x + 1 < BS) blk[y][x + 1] += err;
            if (x + 2 < BS) blk[y][x + 2] += err;
            if (y + 1 < BS) {
                if (x - 1 >= 0) blk[y + 1][x - 1] += err;
                blk[y + 1][x] += err;
                if (x + 1 < BS) blk[y + 1][x + 1] += err;
            }
            if (y + 2 < BS) blk[y + 2][x] += err;
        }
    }

    // ---- store: 2 x b128 per row, the pair covers a wave-contiguous 1 KiB per row ---
    float* obase = out + stripBase + (size_t)t * BS;
    #pragma unroll
    for (int y = 0; y < BS; ++y) {
        float4 lo = make_float4(blk[y][0], blk[y][1], blk[y][2], blk[y][3]);
        float4 hi = make_float4(blk[y][4], blk[y][5], blk[y][6], blk[y][7]);
        *reinterpret_cast<float4*>(obase + (size_t)y * IMW)     = lo;
        *reinterpret_cast<float4*>(obase + (size_t)y * IMW + 4) = hi;
    }
}

extern "C" void kernel_launch(void* const* d_in, const int* in_sizes, int n_in,
                              void* d_out, int out_size, void* d_ws, size_t ws_size,
                              hipStream_t stream) {
    (void)in_sizes; (void)n_in; (void)out_size; (void)d_ws; (void)ws_size;
    const float* in  = (const float*)d_in[0];
    float*       out = (float*)d_out;

    dim3 grid(NIMG * NBY);   // 4096 workgroups, one 64 KiB strip each
    dim3 block(NBX);         // 256 threads = 8 wave32 per workgroup
    hipLaunchKernelGGL(atkinson_dither_kernel, grid, block, 0, stream, in, out);
}